// CenterLoss_57715770523728
// MI455X (gfx1250) — compile-verified
//
#include <hip/hip_runtime.h>

// CenterLoss on gfx1250 (MI455X).
// loss = (1/B) * sum_b || x[b] - centers[labels[b]] ||^2
// B=1024, D=512, C=100000, all f32; labels int32; scalar f32 output.
//
// Memory-bound (4 MiB total traffic). Reduction is done through
// V_WMMA_F32_16X16X4_F32 with B = ones, keeping full f32 precision:
//   D[m][n] += sum_k diff2[m][k]   (row sums replicated across 16 columns)
// Two-pass deterministic reduction (no float atomics).

typedef __attribute__((ext_vector_type(2))) float v2f;
typedef __attribute__((ext_vector_type(8))) float v8f;

#define FEAT_DIM 512
#define ROWS_PER_BLOCK 16   // one WMMA M-tile of batch rows per block
#define WAVES_PER_BLOCK 4   // each wave covers a 128-wide K slice
#define NUM_BLOCKS 64       // 1024 / 16
#define K_PER_WAVE (FEAT_DIM / WAVES_PER_BLOCK)   // 128
#define WMMA_STEPS (K_PER_WAVE / 4)               // 32 steps of K=4

__global__ __launch_bounds__(128) void center_loss_partial(
    const float* __restrict__ x,
    const int*   __restrict__ labels,
    const float* __restrict__ centers,
    float*       __restrict__ partial)
{
    const int lane = threadIdx.x & 31;
    const int wave = threadIdx.x >> 5;

    // A-matrix f32 16x4 layout (ISA 7.12.2):
    //   lanes 0-15:  VGPR0=K0, VGPR1=K1  (M = lane)
    //   lanes 16-31: VGPR0=K2, VGPR1=K3  (M = lane-16)
    const int m    = lane & 15;
    const int koff = (lane >> 4) << 1;           // 0 or 2
    const int row  = blockIdx.x * ROWS_PER_BLOCK + m;

    const int lab = labels[row];                  // gather index for this row
    const float* __restrict__ xr = x       + (size_t)row * FEAT_DIM;
    const float* __restrict__ cr = centers + (size_t)lab * FEAT_DIM;

    const int kbase = wave * K_PER_WAVE;

    v2f bones; bones.x = 1.0f; bones.y = 1.0f;    // B = ones (layout-invariant)
    v8f acc = {};

#pragma unroll 4
    for (int t = 0; t < WMMA_STEPS; ++t) {
        const int k = kbase + (t << 2) + koff;    // even -> 8B aligned
        const v2f xv = *(const v2f*)(xr + k);
        const v2f cv = *(const v2f*)(cr + k);
        const float d0 = xv.x - cv.x;
        const float d1 = xv.y - cv.y;
        v2f a; a.x = d0 * d0; a.y = d1 * d1;
        // D = A x ones + C  -> per-row partial sums accumulate in C/D tile
        acc = __builtin_amdgcn_wmma_f32_16x16x4_f32(
                  false, a, false, bones, (short)0, acc, false, false);
    }

    // Sum the 16x16 accumulator: each (M,N) cell appears exactly once across
    // the 8 VGPRs x 32 lanes; columns replicate row sums 16x -> divide by 16.
    float s = acc[0] + acc[1] + acc[2] + acc[3]
            + acc[4] + acc[5] + acc[6] + acc[7];
#pragma unroll
    for (int off = 16; off > 0; off >>= 1)
        s += __shfl_xor(s, off, 32);

    __shared__ float lds[WAVES_PER_BLOCK];
    if (lane == 0) lds[wave] = s;
    __syncthreads();
    if (threadIdx.x == 0) {
        float blk = (lds[0] + lds[1] + lds[2] + lds[3]) * (1.0f / 16.0f);
        partial[blockIdx.x] = blk;
    }
}

__global__ __launch_bounds__(32) void center_loss_final(
    const float* __restrict__ partial,
    float*       __restrict__ out)
{
    // 64 partials, one wave: two per lane, then shuffle tree. Deterministic.
    float s = partial[threadIdx.x] + partial[threadIdx.x + 32];
#pragma unroll
    for (int off = 16; off > 0; off >>= 1)
        s += __shfl_xor(s, off, 32);
    if (threadIdx.x == 0)
        out[0] = s * (1.0f / 1024.0f);
}

extern "C" void kernel_launch(void* const* d_in, const int* in_sizes, int n_in,
                              void* d_out, int out_size, void* d_ws, size_t ws_size,
                              hipStream_t stream) {
    const float* x       = (const float*)d_in[0];
    const int*   labels  = (const int*)  d_in[1];
    const float* centers = (const float*)d_in[2];
    float*       out     = (float*)d_out;
    float*       partial = (float*)d_ws;          // 64 floats of scratch

    center_loss_partial<<<NUM_BLOCKS, ROWS_PER_BLOCK * 8, 0, stream>>>(
        x, labels, centers, partial);
    center_loss_final<<<1, 32, 0, stream>>>(partial, out);
}